// RxaFilterEnsemble_8091718385913
// MI455X (gfx1250) — compile-verified
//
#include <hip/hip_runtime.h>
#include <cstdint>

// Problem constants (match reference): x[B,S,C,2], w_real/w_imag[C,S]
#define Bsz     256
#define Ssz     4096
#define Csz     64
#define NSLOT   128                  // C*2 floats per (b,s) row = 512 bytes
#define SCHUNK  64                   // s-rows per LDS tile
#define TILE_FLOATS (SCHUNK * NSLOT) // 8192 floats = 32 KB
#define BGROUP  16                   // batches processed per block
#define NCHUNK  (Ssz / SCHUNK)       // 64
#define THREADS 256                  // 8 waves (wave32)

// ---- CDNA5 async global->LDS copy (ASYNCcnt path) ----------------------
#if defined(__has_builtin)
# if __has_builtin(__builtin_amdgcn_global_load_async_to_lds_b128)
#  define ASYNC_BUILTIN 1
# endif
# if __has_builtin(__builtin_amdgcn_s_wait_asynccnt)
#  define WAITA_BUILTIN 1
# endif
#endif

typedef int v4i __attribute__((vector_size(16)));
typedef __attribute__((address_space(1))) v4i* gv4i_p;   // global
typedef __attribute__((address_space(3))) v4i* sv4i_p;   // LDS

__device__ __forceinline__ void async_b128(void* lds_dst, const float* gsrc) {
#ifdef ASYNC_BUILTIN
  __builtin_amdgcn_global_load_async_to_lds_b128(
      (gv4i_p)(void*)gsrc,   // generic -> AS1, const cast via (void*)
      (sv4i_p)lds_dst,       // generic -> AS3 addrspacecast
      /*imm offset*/ 0, /*cpol*/ 0);
#else
  unsigned off = (unsigned)(uintptr_t)(__attribute__((address_space(3))) void*)lds_dst;
  asm volatile("global_load_async_to_lds_b128 %0, %1, off"
               :: "v"(off), "v"(gsrc)
               : "memory");
#endif
}

template <int N>
__device__ __forceinline__ void wait_async() {
#ifdef WAITA_BUILTIN
  __builtin_amdgcn_s_wait_asynccnt(N);
#else
  asm volatile("s_wait_asynccnt %0" :: "i"(N) : "memory");
#endif
}

// ---- main kernel (defined first so it leads the disassembly) -----------
__global__ __launch_bounds__(THREADS)
void rxa_filter_kernel(const float* __restrict__ x,
                       const float* __restrict__ wr,
                       const float* __restrict__ wi,
                       float* __restrict__ out) {
  __shared__ float xs[2][TILE_FLOATS];   // 64 KB double buffer

  const int tid   = threadIdx.x;
  const int slot  = tid & (NSLOT - 1);   // c*2 + comp
  const int shalf = tid >> 7;            // which half of the s-chunk
  const int c     = slot >> 1;
  const int comp  = slot & 1;

  const int chunk = blockIdx.x % NCHUNK;
  const int bg    = blockIdx.x / NCHUNK;
  const int b0    = bg * BGROUP;
  const int s0    = chunk * SCHUNK;

  // Preload this thread's 32 weights into VGPRs (contiguous along s, 128B aligned).
  const float* wsrc = (comp ? wi : wr) + (size_t)c * Ssz + s0 + shalf * 32;
  float w[32];
  {
    const float4* w4 = reinterpret_cast<const float4*>(wsrc);
#pragma unroll
    for (int q = 0; q < 8; ++q) {
      float4 t = w4[q];
      w[4*q+0] = t.x; w[4*q+1] = t.y; w[4*q+2] = t.z; w[4*q+3] = t.w;
    }
  }

  // Prologue: async-fill buffer 0 with batch b0 (tile is 32 KB contiguous).
  {
    const float* src = x + ((size_t)b0 * Ssz + s0) * NSLOT;
#pragma unroll
    for (int k = 0; k < 8; ++k)
      async_b128(&xs[0][k * 1024 + tid * 4], src + k * 1024 + tid * 4);
  }

  for (int i = 0; i < BGROUP; ++i) {
    const int buf = i & 1;

    if (i + 1 < BGROUP) {
      // Prefetch next batch's tile into the other buffer (safe: trailing
      // barrier of iteration i-1 guarantees everyone finished reading it).
      const float* src = x + ((size_t)(b0 + i + 1) * Ssz + s0) * NSLOT;
#pragma unroll
      for (int k = 0; k < 8; ++k)
        async_b128(&xs[buf ^ 1][k * 1024 + tid * 4], src + k * 1024 + tid * 4);
      wait_async<8>();   // current tile's 8 asyncs done; prefetch stays in flight
    } else {
      wait_async<0>();
    }
    __syncthreads();     // all waves' fills for buf are complete

    // 32 conflict-free LDS loads (lane-consecutive 4B) + 32 FMAs.
    float acc = 0.0f;
    const float* col = &xs[buf][(shalf * 32) * NSLOT + slot];
#pragma unroll
    for (int j = 0; j < 32; ++j)
      acc = fmaf(col[j * NSLOT], w[j], acc);

    // Native f32 global atomic: cross-(chunk,shalf) reduction.
    unsafeAtomicAdd(out + (size_t)(b0 + i) * NSLOT + slot, acc);
    __syncthreads();     // protect buf from being refilled while others read
  }
}

// ---- output zero-init --------------------------------------------------
__global__ void rxa_zero_kernel(float* __restrict__ out, int n) {
  int i = blockIdx.x * blockDim.x + threadIdx.x;
  if (i < n) out[i] = 0.0f;
}

// ---- entry point -------------------------------------------------------
extern "C" void kernel_launch(void* const* d_in, const int* in_sizes, int n_in,
                              void* d_out, int out_size, void* d_ws, size_t ws_size,
                              hipStream_t stream) {
  (void)in_sizes; (void)n_in; (void)d_ws; (void)ws_size;
  const float* x  = (const float*)d_in[0];
  const float* wr = (const float*)d_in[1];
  const float* wi = (const float*)d_in[2];
  float* out = (float*)d_out;

  rxa_zero_kernel<<<(out_size + 255) / 256, 256, 0, stream>>>(out, out_size);

  dim3 grid(NCHUNK * (Bsz / BGROUP));   // 64 * 16 = 1024 blocks
  rxa_filter_kernel<<<grid, THREADS, 0, stream>>>(x, wr, wi, out);
}